// LocalAttention_20736102105806
// MI455X (gfx1250) — compile-verified
//
#include <hip/hip_runtime.h>

typedef float v2f __attribute__((ext_vector_type(2)));
typedef float v8f __attribute__((ext_vector_type(8)));

#define CCH  64
#define HH   256
#define WW   256
#define HL   128
#define WL   128

__device__ __forceinline__ int clampi(int v, int lo, int hi) {
    return v < lo ? lo : (v > hi ? hi : v);
}

// Bilinear (align_corners=True) sample of lr_up at (oy,ox), oy/ox in [0,255].
__device__ __forceinline__ float lrup_plane(const float* __restrict__ p, int oy, int ox) {
    const float sc = 127.0f / 255.0f;
    float fy = oy * sc, fx = ox * sc;
    int y0 = (int)fy, x0 = (int)fx;
    float wy = fy - (float)y0, wx = fx - (float)x0;
    int y1 = y0 + 1 < HL ? y0 + 1 : HL - 1;
    int x1 = x0 + 1 < WL ? x0 + 1 : WL - 1;
    float v00 = p[y0 * WL + x0], v01 = p[y0 * WL + x1];
    float v10 = p[y1 * WL + x0], v11 = p[y1 * WL + x1];
    float t0 = v00 + wx * (v01 - v00);
    float t1 = v10 + wx * (v11 - v10);
    return t0 + wy * (t1 - t0);
}

// Q = depthwise 3x3 over lr_up ('SAME' zero pad). w9 lives in VGPRs (caller
// preloads). Pixel loads clamped + unconditionally consumed; the zero-pad
// select is applied to the *register* weight so loads cannot be predicated.
__device__ __forceinline__ float qconv_w(const float* __restrict__ lrp,
                                         const float* w9, float bias,
                                         int y, int x) {
    float s = bias;
#pragma unroll
    for (int dy = -1; dy <= 1; ++dy) {
        int yy = y + dy; int yc = clampi(yy, 0, HH - 1); bool yok = (yy == yc);
#pragma unroll
        for (int dx = -1; dx <= 1; ++dx) {
            int xx = x + dx; int xc = clampi(xx, 0, WW - 1);
            float px = lrup_plane(lrp, yc, xc);                 // unconditional
            float wm = (yok && xx == xc) ? w9[(dy + 1) * 3 + (dx + 1)] : 0.0f;
            s += wm * px;
        }
    }
    return s;
}

// K/V = depthwise 3x3 over an hr_feat plane; result forced to 0 when the
// center (y,x) is outside the image (multiplicative mask, unsinkable).
__device__ __forceinline__ float hrconv_w(const float* __restrict__ hp,
                                          const float* w9, float bias,
                                          int y, int x) {
    float s = bias;
#pragma unroll
    for (int dy = -1; dy <= 1; ++dy) {
        int yy = y + dy; int yc = clampi(yy, 0, HH - 1); bool yok = (yy == yc);
#pragma unroll
        for (int dx = -1; dx <= 1; ++dx) {
            int xx = x + dx; int xc = clampi(xx, 0, WW - 1);
            float px = hp[yc * WW + xc];                        // unconditional load
            float wm = (yok && xx == xc) ? w9[(dy + 1) * 3 + (dx + 1)] : 0.0f;
            s += wm * px;
        }
    }
    float cmask = (((unsigned)y < (unsigned)HH) && ((unsigned)x < (unsigned)WW))
                      ? 1.0f : 0.0f;
    return s * cmask;
}

__device__ __forceinline__ void wave_lds_fence() {
    // Per-wave LDS producer->consumer ordering: DS pipe is in-order per wave;
    // the fence stops compiler reordering and drains outstanding DS ops.
    asm volatile("s_wait_dscnt 0x0" ::: "memory");
}

__global__ __launch_bounds__(128)
void sapa_local_attn_kernel(const float* __restrict__ hr, const float* __restrict__ lr,
                            const float* __restrict__ qw_g, const float* __restrict__ qb_g,
                            const float* __restrict__ kw_g, const float* __restrict__ kb_g,
                            const float* __restrict__ vw_g, const float* __restrict__ vb_g,
                            float* __restrict__ out) {
    __shared__ float s_qw[CCH * 9], s_kw[CCH * 9], s_vw[CCH * 9];
    __shared__ float s_qb[CCH], s_kb[CCH], s_vb[CCH];
    __shared__ float s_q[4][CCH * 16];     // per-wave Q tile, [ch][m]
    __shared__ float s_band[4][16 * 32];   // per-wave: S band tile / W2 / reductions
    __shared__ float s_P[4][16 * 52];      // per-wave: 49 softmax weights per pixel

    for (int i = threadIdx.x; i < CCH * 9; i += blockDim.x) {
        s_qw[i] = qw_g[i]; s_kw[i] = kw_g[i]; s_vw[i] = vw_g[i];
    }
    for (int i = threadIdx.x; i < CCH; i += blockDim.x) {
        s_qb[i] = qb_g[i]; s_kb[i] = kb_g[i]; s_vb[i] = vb_g[i];
    }
    __syncthreads();   // weights shared block-wide; only block barrier needed

    const int lane = threadIdx.x & 31;
    const int wv   = threadIdx.x >> 5;
    const int tile = blockIdx.x * 4 + wv;          // 16384 wave-tiles
    const int tw   = tile & 15;
    const int h    = (tile >> 4) & 255;
    const int n    = tile >> 12;
    const int w0   = tw * 16;

    float* Qw   = s_q[wv];
    float* band = s_band[wv];
    float* P    = s_P[wv];

    const int m16  = lane & 15;
    const int ksub = (lane >> 4) << 1;   // lanes 0-15: K={0,1}; lanes 16-31: K={2,3}
    const int Mb   = (lane >> 4) * 8;

    const float* lrbase = lr + (size_t)n * CCH * HL * WL;
    const float* hrbase = hr + (size_t)n * CCH * HH * WW;

    // ---- Q tile (64 ch x 16 px) computed cooperatively into per-wave LDS ----
#pragma unroll 1
    for (int it = 0; it < 32; ++it) {              // 1024 values / 32 lanes
        int idx = lane + it * 32;
        int ch = idx >> 4, m = idx & 15;
        float wq[9];
#pragma unroll
        for (int t9 = 0; t9 < 9; ++t9) wq[t9] = s_qw[ch * 9 + t9];
        Qw[idx] = qconv_w(lrbase + (size_t)ch * HL * WL, wq, s_qb[ch], h, w0 + m);
    }
    wave_lds_fence();

    // ---- Stage 1+2: per di, S = Q·K^T (fp32 WMMA), band-extract into P ----
#pragma unroll 1
    for (int di = 0; di < 7; ++di) {
        int y = h + di - 3;
        // CDNA5 prefetch: pull next di's key row into cache ahead of the convs.
        if (di < 6) {
            int yn = clampi(y + 1, 0, HH - 1);
            __builtin_prefetch(hrbase + (size_t)(lane * 2) * HH * WW
                                      + (size_t)yn * WW + w0, 0, 3);
        }
        v8f S0 = {}, S1 = {};
#pragma unroll 2
        for (int ci = 0; ci < 16; ++ci) {
            int c0 = 4 * ci + ksub;
            const float* kp0 = hrbase + (size_t)c0 * HH * WW;
            const float* kp1 = kp0 + HH * WW;
            float wk0[9], wk1[9];
#pragma unroll
            for (int t9 = 0; t9 < 9; ++t9) {       // unconditional weight loads
                wk0[t9] = s_kw[c0 * 9 + t9];
                wk1[t9] = s_kw[c0 * 9 + 9 + t9];
            }
            float kb0 = s_kb[c0], kb1 = s_kb[c0 + 1];
            v2f a;                                 // A fragment from LDS Q tile
            a.x = Qw[c0 * 16 + m16];
            a.y = Qw[(c0 + 1) * 16 + m16];
            int x0 = w0 - 3 + m16;
            v2f b0, b1;
            b0.x = hrconv_w(kp0, wk0, kb0, y, x0);
            b0.y = hrconv_w(kp1, wk1, kb1, y, x0);
            b1.x = hrconv_w(kp0, wk0, kb0, y, x0 + 16);
            b1.y = hrconv_w(kp1, wk1, kb1, y, x0 + 16);
            S0 = __builtin_amdgcn_wmma_f32_16x16x4_f32(false, a, false, b0,
                                                       (short)0, S0, false, false);
            S1 = __builtin_amdgcn_wmma_f32_16x16x4_f32(false, a, false, b1,
                                                       (short)0, S1, false, false);
        }
        // spill S to per-wave LDS band [M][N], stride 32 (cols 22..31 unused)
#pragma unroll
        for (int r = 0; r < 8; ++r) {
            band[(Mb + r) * 32 + m16]      = S0[r];
            band[(Mb + r) * 32 + 16 + m16] = S1[r];
        }
        wave_lds_fence();
        // uniform-trip band extraction (idx clamp -> benign duplicate write)
#pragma unroll 1
        for (int it = 0; it < 4; ++it) {
            int idx = lane + it * 32;
            idx = idx < 112 ? idx : 111;
            int m = idx / 7, dj = idx - m * 7;
            P[m * 52 + di * 7 + dj] = band[m * 32 + m + dj];   // weight[m, di*7+dj]
        }
        wave_lds_fence();
    }

    // ---- Softmax over 49 taps: two lanes per pixel (uniform EXEC) ----
    {
        float* Pp = P + m16 * 52;
        const int half = lane >> 4;          // 0: taps 0..24, 1: taps 25..48
        const int i0   = half ? 25 : 0;
        const int cnt  = half ? 24 : 25;
        float pm = -3.4e38f;
#pragma unroll 1
        for (int i = 0; i < 25; ++i) {
            int t = i0 + i; t = t < 49 ? t : 0;
            float v = Pp[t];
            pm = (i < cnt) ? fmaxf(pm, v) : pm;
        }
        band[lane] = pm;                     // exchange between halves
        wave_lds_fence();
        float mx = fmaxf(pm, band[lane ^ 16]);
        wave_lds_fence();
        float psum = 0.0f;
#pragma unroll 1
        for (int i = 0; i < 25; ++i) {
            int t = i0 + i; t = t < 49 ? t : 0;
            float e = __expf(Pp[t] - mx);
            bool ok = (i < cnt);
            if (ok) Pp[t] = e;
            psum += ok ? e : 0.0f;
        }
        wave_lds_fence();
        band[lane] = psum;
        wave_lds_fence();
        float inv = 1.0f / (psum + band[lane ^ 16]);
#pragma unroll 1
        for (int i = 0; i < 25; ++i) {
            int t = i0 + i; t = t < 49 ? t : 0;
            if (i < cnt) Pp[t] *= inv;
        }
        wave_lds_fence();
    }

    // ---- Stage 3: attn[16ch x 16px] += V · banded(W) via fp32 WMMA ----
    // V conv weights depend only on ct -> preload all 4x9 into VGPRs once.
    float wv9[4][9], vb4[4];
#pragma unroll
    for (int ct = 0; ct < 4; ++ct) {
        int ch = ct * 16 + m16;
#pragma unroll
        for (int t9 = 0; t9 < 9; ++t9) wv9[ct][t9] = s_vw[ch * 9 + t9];
        vb4[ct] = s_vb[ch];
    }

    v8f D[4];
#pragma unroll
    for (int ct = 0; ct < 4; ++ct) D[ct] = {};

#pragma unroll 1
    for (int di = 0; di < 7; ++di) {
        int y = h + di - 3;
        if (di < 6) {   // prefetch next value row
            int yn = clampi(y + 1, 0, HH - 1);
            __builtin_prefetch(hrbase + (size_t)(lane * 2) * HH * WW
                                      + (size_t)yn * WW + w0, 0, 3);
        }
#pragma unroll 1
        for (int it = 0; it < 12; ++it) band[lane + it * 32] = 0.0f;   // W2 := 0
        wave_lds_fence();
#pragma unroll 1
        for (int it = 0; it < 4; ++it) {     // scatter the band weights
            int idx = lane + it * 32;
            idx = idx < 112 ? idx : 111;
            int m = idx / 7, dj = idx - m * 7;
            band[(m + dj) * 16 + m] = P[m * 52 + di * 7 + dj];   // W2[j'][m]
        }
        wave_lds_fence();
#pragma unroll
        for (int ct = 0; ct < 4; ++ct) {
            int ch = ct * 16 + m16;
            const float* vp = hrbase + (size_t)ch * HH * WW;
#pragma unroll 2
            for (int k4 = 0; k4 < 24; k4 += 4) {
                int jj = k4 + ksub;
                int x  = w0 - 3 + jj;
                v2f a, b;
                a.x = hrconv_w(vp, wv9[ct], vb4[ct], y, x);    // A = V[16ch x 4j']
                a.y = hrconv_w(vp, wv9[ct], vb4[ct], y, x + 1);
                b.x = band[jj * 16 + m16];                     // B = W2[4j' x 16px]
                b.y = band[(jj + 1) * 16 + m16];
                D[ct] = __builtin_amdgcn_wmma_f32_16x16x4_f32(false, a, false, b,
                                                              (short)0, D[ct], false, false);
            }
        }
        wave_lds_fence();   // before next di rewrites W2
    }

    // ---- out = lr_up + attn ----
#pragma unroll
    for (int ct = 0; ct < 4; ++ct) {
#pragma unroll
        for (int r = 0; r < 8; ++r) {
            int ch = ct * 16 + Mb + r;
            float lu = lrup_plane(lrbase + (size_t)ch * HL * WL, h, w0 + m16);
            out[((size_t)(n * CCH + ch) * HH + h) * WW + w0 + m16] = lu + D[ct][r];
        }
    }
}

extern "C" void kernel_launch(void* const* d_in, const int* in_sizes, int n_in,
                              void* d_out, int out_size, void* d_ws, size_t ws_size,
                              hipStream_t stream) {
    const float* hr = (const float*)d_in[0];
    const float* lr = (const float*)d_in[1];
    const float* qw = (const float*)d_in[2];
    const float* qb = (const float*)d_in[3];
    const float* kw = (const float*)d_in[4];
    const float* kb = (const float*)d_in[5];
    const float* vw = (const float*)d_in[6];
    const float* vb = (const float*)d_in[7];
    float* out = (float*)d_out;

    dim3 grid(4096), block(128);   // 16384 wave-tiles, 4 waves/block
    hipLaunchKernelGGL(sapa_local_attn_kernel, grid, block, 0, stream,
                       hr, lr, qw, qb, kw, kb, vw, vb, out);
}